// GCN_18760417149681
// MI455X (gfx1250) — compile-verified
//
#include <hip/hip_runtime.h>

// ---------------------------------------------------------------------------
// GraphSAGE forward on MI455X (gfx1250, wave32).
//   GEMMs   -> V_WMMA_F32_16X16X4_F32, compile-time K/M, 4 col-tiles per wave,
//              weights pre-packed into fragment order (one b64 per B fragment)
//   Scatter -> global_atomic_add_f32, coalesced over the feature dimension
// ---------------------------------------------------------------------------

typedef __attribute__((ext_vector_type(2))) float v2f;
typedef __attribute__((ext_vector_type(8))) float v8f;

#define N_NODES 100000
#define N_EDGES 1600000

// ---------------- zero workspace (atomics accumulate, so zero each call) ----
__global__ void gcn_zero_kernel(float* __restrict__ p, long long n) {
    long long i = (long long)blockIdx.x * blockDim.x + threadIdx.x;
    long long stride = (long long)gridDim.x * blockDim.x;
    for (; i < n; i += stride) p[i] = 0.0f;
}

// ---------------- edge scatter-add: agg[dst] += feat[src]; deg[dst] += 1 ----
// one thread per (edge, feature); F = 1<<logF (64 or 128)
__global__ void gcn_scatter_kernel(const float* __restrict__ feat,
                                   const int* __restrict__ src,
                                   const int* __restrict__ dst,
                                   float* __restrict__ agg,
                                   float* __restrict__ deg,   // nullptr on 2nd layer
                                   int nEdges, int logF) {
    long long t = (long long)blockIdx.x * blockDim.x + threadIdx.x;
    int e = (int)(t >> logF);
    if (e >= nEdges) return;
    int f = (int)(t & ((1 << logF) - 1));
    int s = src[e];
    int d = dst[e];
    atomicAdd(&agg[((long long)d << logF) + f], feat[((long long)s << logF) + f]);
    if (deg != nullptr && f == 0) atomicAdd(&deg[d], 1.0f);
}

// ---------------- mean = agg / max(deg, 1) in place ------------------------
__global__ void gcn_mean_kernel(float* __restrict__ agg,
                                const float* __restrict__ deg,
                                long long n, int logF) {
    long long i = (long long)blockIdx.x * blockDim.x + threadIdx.x;
    long long stride = (long long)gridDim.x * blockDim.x;
    for (; i < n; i += stride) {
        float dv = deg[i >> logF];
        agg[i] = agg[i] / fmaxf(dv, 1.0f);
    }
}

// ---------------- pack W[K][M] into WMMA B-fragment order -------------------
// For k-group g=k/4 and lane-half h, a lane's B fragment is
// {W[4g+2h][n], W[4g+2h+1][n]}. Store those pairs contiguously:
//   Wp[((g*2+h)*M + n)*2 + {0,1}]
// so the GEMM loads one 8-byte b64 per fragment instead of two strided b32s.
__global__ void gcn_pack_w_kernel(const float* __restrict__ W,
                                  float* __restrict__ Wp,
                                  int K, int M) {
    int i = blockIdx.x * blockDim.x + threadIdx.x;   // over (K/2)*M pairs
    int total = (K / 2) * M;
    if (i >= total) return;
    int pairIdx = i / M;                 // g*2 + h
    int n = i - pairIdx * M;
    int g = pairIdx >> 1;
    int h = pairIdx & 1;
    int k = 4 * g + 2 * h;
    Wp[(long long)i * 2 + 0] = W[(long long)k * M + n];
    Wp[(long long)i * 2 + 1] = W[(long long)(k + 1) * M + n];
}

// ---------------- WMMA GEMM:  C = A1@W1p (+ A2@W2p) + bias -----------------
// One wave computes a 16 x (16*TN) strip of C via V_WMMA_F32_16X16X4_F32,
// reusing one A fragment across TN column tiles.
//   A (16x4 f32): lane<16 holds {A[r][k], A[r][k+1]}, lanes 16..31 hold k+2,k+3
//   B (4x16 f32): v0 = rows k / k+2 across lane halves, v1 = rows k+1 / k+3
//   C/D (16x16) : VGPR v -> row (v + 8*laneHalf), col = lane&15
// W1p/W2p are in packed fragment order (see gcn_pack_w_kernel): float index
//   k*M + 32*t relative to (h*2M + 2*col) base -> immediate-offset b64 loads.
// N multiple of 16 -> no guards; EXEC stays all-ones as WMMA requires.
template <int K, int M, int TN, bool DUAL>
__global__ void gcn_wmma_gemm_kernel(const float* __restrict__ A1,
                                     const float* __restrict__ W1p,
                                     const float* __restrict__ A2,
                                     const float* __restrict__ W2p,
                                     const float* __restrict__ bias,
                                     float* __restrict__ C) {
    const int lane   = threadIdx.x;        // 0..31
    const int half   = lane >> 4;          // 0 or 1
    const int lane16 = lane & 15;
    const long long rowBase = (long long)blockIdx.x * 16;
    const int colBase = threadIdx.y * (16 * TN);       // first output column
    const long long r = rowBase + lane16;              // A row for this lane

    const float* a1p = A1 + r * K + 2 * half;
    const float* w1p = W1p + half * 2 * M + 2 * (colBase + lane16);
    const float* a2p = DUAL ? (A2 + r * K + 2 * half) : nullptr;
    const float* w2p = DUAL ? (W2p + half * 2 * M + 2 * (colBase + lane16)) : nullptr;

    v8f acc[TN];
    #pragma unroll
    for (int t = 0; t < TN; ++t) acc[t] = (v8f){};

    #pragma unroll 4
    for (int k = 0; k < K; k += 4) {
        v2f a1 = *(const v2f*)(a1p + k);               // 8B-aligned b64 load
        v2f a2;
        if (DUAL) a2 = *(const v2f*)(a2p + k);
        #pragma unroll
        for (int t = 0; t < TN; ++t) {
            v2f b1 = *(const v2f*)(w1p + k * M + 32 * t);   // one b64 per fragment
            acc[t] = __builtin_amdgcn_wmma_f32_16x16x4_f32(
                /*neg_a=*/false, a1, /*neg_b=*/false, b1,
                /*c_mod=*/(short)0, acc[t], /*reuse_a=*/false, /*reuse_b=*/false);
            if (DUAL) {
                v2f b2 = *(const v2f*)(w2p + k * M + 32 * t);
                acc[t] = __builtin_amdgcn_wmma_f32_16x16x4_f32(
                    false, a2, false, b2, (short)0, acc[t], false, false);
            }
        }
    }

    #pragma unroll
    for (int t = 0; t < TN; ++t) {
        const int col = colBase + 16 * t + lane16;
        const float bv = bias[col];
        #pragma unroll
        for (int v = 0; v < 8; ++v) {
            long long row = rowBase + v + 8 * half;
            C[row * M + col] = acc[t][v] + bv;
        }
    }
}

// ---------------- head: out[i] = h3[i,:32] . Wfc2 + bfc2 -------------------
__global__ void gcn_head_kernel(const float* __restrict__ h3,
                                const float* __restrict__ w,
                                const float* __restrict__ b,
                                float* __restrict__ out, int n) {
    int i = blockIdx.x * blockDim.x + threadIdx.x;
    if (i >= n) return;
    const float* row = h3 + (long long)i * 32;
    float s = b[0];
    #pragma unroll
    for (int c = 0; c < 32; ++c) s += row[c] * w[c];
    out[i] = s;
}

// ---------------------------------------------------------------------------
extern "C" void kernel_launch(void* const* d_in, const int* in_sizes, int n_in,
                              void* d_out, int out_size, void* d_ws, size_t ws_size,
                              hipStream_t stream) {
    const float* x    = (const float*)d_in[0];
    const int*   eidx = (const int*)d_in[1];          // [2, E] flattened
    const int*   src  = eidx;
    const int*   dst  = eidx + N_EDGES;
    // d_in[2] = edge_weight (unused by reference mean aggregation)
    const float* Wl1  = (const float*)d_in[3];
    const float* Wr1  = (const float*)d_in[4];
    const float* b1   = (const float*)d_in[5];
    const float* Wl2  = (const float*)d_in[6];
    const float* Wr2  = (const float*)d_in[7];
    const float* b2   = (const float*)d_in[8];
    const float* Wfc1 = (const float*)d_in[9];
    const float* bfc1 = (const float*)d_in[10];
    const float* Wfc2 = (const float*)d_in[11];
    const float* bfc2 = (const float*)d_in[12];
    float* out = (float*)d_out;

    // workspace layout (floats)
    float* agg  = (float*)d_ws;                        // N*128 (64 used in L1)
    float* deg  = agg + (size_t)N_NODES * 128;         // N
    float* h1   = deg + N_NODES;                       // N*128
    float* h2   = h1 + (size_t)N_NODES * 128;          // N*128
    float* h3   = h2 + (size_t)N_NODES * 128;          // N*32
    float* Wl1p = h3 + (size_t)N_NODES * 32;           // 64*128
    float* Wr1p = Wl1p + 64 * 128;                     // 64*128
    float* Wl2p = Wr1p + 64 * 128;                     // 128*128
    float* Wr2p = Wl2p + 128 * 128;                    // 128*128
    float* Wfc1p = Wr2p + 128 * 128;                   // 128*32

    const int T = 256;

    // ---- pack weights into WMMA fragment order (tiny, once per launch) ----
    gcn_pack_w_kernel<<<(32 * 128 + T - 1) / T, T, 0, stream>>>(Wl1, Wl1p, 64, 128);
    gcn_pack_w_kernel<<<(32 * 128 + T - 1) / T, T, 0, stream>>>(Wr1, Wr1p, 64, 128);
    gcn_pack_w_kernel<<<(64 * 128 + T - 1) / T, T, 0, stream>>>(Wl2, Wl2p, 128, 128);
    gcn_pack_w_kernel<<<(64 * 128 + T - 1) / T, T, 0, stream>>>(Wr2, Wr2p, 128, 128);
    gcn_pack_w_kernel<<<(64 * 32 + T - 1) / T, T, 0, stream>>>(Wfc1, Wfc1p, 128, 32);

    // ---- layer 1: mean-aggregate x (F=64), then h1 = mean@Wl1 + x@Wr1 + b1
    gcn_zero_kernel<<<2048, T, 0, stream>>>(agg, (long long)N_NODES * 64);
    gcn_zero_kernel<<<256, T, 0, stream>>>(deg, (long long)N_NODES);
    {
        long long nt = (long long)N_EDGES * 64;
        gcn_scatter_kernel<<<(int)((nt + T - 1) / T), T, 0, stream>>>(
            x, src, dst, agg, deg, N_EDGES, 6);
        gcn_mean_kernel<<<2048, T, 0, stream>>>(agg, deg, (long long)N_NODES * 64, 6);
    }
    {
        dim3 blk(32, 128 / (16 * 4));                  // 2 waves x 4 col tiles
        gcn_wmma_gemm_kernel<64, 128, 4, true><<<N_NODES / 16, blk, 0, stream>>>(
            agg, Wl1p, x, Wr1p, b1, h1);
    }

    // ---- layer 2: mean-aggregate h1 (F=128), then h2 = mean@Wl2 + h1@Wr2 + b2
    gcn_zero_kernel<<<4096, T, 0, stream>>>(agg, (long long)N_NODES * 128);
    {
        long long nt = (long long)N_EDGES * 128;
        gcn_scatter_kernel<<<(int)((nt + T - 1) / T), T, 0, stream>>>(
            h1, src, dst, agg, /*deg=*/nullptr, N_EDGES, 7);
        gcn_mean_kernel<<<4096, T, 0, stream>>>(agg, deg, (long long)N_NODES * 128, 7);
    }
    {
        dim3 blk(32, 128 / (16 * 4));
        gcn_wmma_gemm_kernel<128, 128, 4, true><<<N_NODES / 16, blk, 0, stream>>>(
            agg, Wl2p, h1, Wr2p, b2, h2);
    }

    // ---- fc head: h3 = h2@Wfc1 + bfc1 (K=128, M=32)
    {
        dim3 blk(32, 1);                               // 1 wave x 2 col tiles
        gcn_wmma_gemm_kernel<128, 32, 2, false><<<N_NODES / 16, blk, 0, stream>>>(
            h2, Wfc1p, nullptr, nullptr, bfc1, h3);
    }
    gcn_head_kernel<<<(N_NODES + T - 1) / T, T, 0, stream>>>(h3, Wfc2, bfc2, out, N_NODES);
}